// EnergyFunction_29308856827972
// MI455X (gfx1250) — compile-verified
//
#include <hip/hip_runtime.h>
#include <hip/hip_bf16.h>
#include <math.h>

typedef float v2f __attribute__((ext_vector_type(2)));
typedef float v8f __attribute__((ext_vector_type(8)));
typedef unsigned int u32x4 __attribute__((ext_vector_type(4)));
typedef int i32x4 __attribute__((ext_vector_type(4)));
typedef int i32x8 __attribute__((ext_vector_type(8)));

#define S_LEN   4096
#define F_DIM   512
#define D_EN    64
#define KROW_PAD 66   // 64 f32 + 2-dword TDM pad per row -> no 64-bank alignment

// ---------------------------------------------------------------------------
// Kernel 0: transpose Wq/Wk [512,64] -> WqT/WkT [64,512]
// ---------------------------------------------------------------------------
__global__ void ef_transpose_kernel(const float* __restrict__ Wq,
                                    const float* __restrict__ Wk,
                                    float* __restrict__ WqT,
                                    float* __restrict__ WkT) {
    int idx = blockIdx.x * blockDim.x + threadIdx.x;   // 0 .. 512*64-1
    int n = idx & 63;
    int d = idx >> 6;
    WqT[n * F_DIM + d] = Wq[idx];
    WkT[n * F_DIM + d] = Wk[idx];
}

// ---------------------------------------------------------------------------
// Kernel 1: Q = features @ Wq, K = features @ Wk  (WMMA f32 16x16x4)
// One wave computes a [16 x 64] tile of both Q and K.
// ---------------------------------------------------------------------------
__global__ void ef_proj_kernel(const float* __restrict__ feats,   // [16384,512]
                               const float* __restrict__ WqT,     // [64,512]
                               const float* __restrict__ WkT,     // [64,512]
                               float* __restrict__ Qout,          // [16384,64]
                               float* __restrict__ Kout) {        // [16384,64]
    const int wid  = threadIdx.x >> 5;
    const int lane = threadIdx.x & 31;
    const int lo   = lane & 15;
    const int hi   = lane >> 4;
    const int rowTile = blockIdx.x * 8 + wid;          // 1024 tiles total
    const int row0 = rowTile * 16;

    const float* fRow = feats + (size_t)(row0 + lo) * F_DIM;

    v8f q0 = {}, q1 = {}, q2 = {}, q3 = {};
    v8f k0 = {}, k1 = {}, k2 = {}, k3 = {};

#pragma unroll 4
    for (int kk = 0; kk < F_DIM / 4; ++kk) {
        const int d = kk * 4 + hi * 2;
        v2f a = *(const v2f*)(fRow + d);

        v2f bq0 = *(const v2f*)(WqT + (size_t)(0 * 16 + lo) * F_DIM + d);
        v2f bq1 = *(const v2f*)(WqT + (size_t)(1 * 16 + lo) * F_DIM + d);
        v2f bq2 = *(const v2f*)(WqT + (size_t)(2 * 16 + lo) * F_DIM + d);
        v2f bq3 = *(const v2f*)(WqT + (size_t)(3 * 16 + lo) * F_DIM + d);
        v2f bk0 = *(const v2f*)(WkT + (size_t)(0 * 16 + lo) * F_DIM + d);
        v2f bk1 = *(const v2f*)(WkT + (size_t)(1 * 16 + lo) * F_DIM + d);
        v2f bk2 = *(const v2f*)(WkT + (size_t)(2 * 16 + lo) * F_DIM + d);
        v2f bk3 = *(const v2f*)(WkT + (size_t)(3 * 16 + lo) * F_DIM + d);

        q0 = __builtin_amdgcn_wmma_f32_16x16x4_f32(false, a, false, bq0, (short)0, q0, false, false);
        q1 = __builtin_amdgcn_wmma_f32_16x16x4_f32(false, a, false, bq1, (short)0, q1, false, false);
        q2 = __builtin_amdgcn_wmma_f32_16x16x4_f32(false, a, false, bq2, (short)0, q2, false, false);
        q3 = __builtin_amdgcn_wmma_f32_16x16x4_f32(false, a, false, bq3, (short)0, q3, false, false);
        k0 = __builtin_amdgcn_wmma_f32_16x16x4_f32(false, a, false, bk0, (short)0, k0, false, false);
        k1 = __builtin_amdgcn_wmma_f32_16x16x4_f32(false, a, false, bk1, (short)0, k1, false, false);
        k2 = __builtin_amdgcn_wmma_f32_16x16x4_f32(false, a, false, bk2, (short)0, k2, false, false);
        k3 = __builtin_amdgcn_wmma_f32_16x16x4_f32(false, a, false, bk3, (short)0, k3, false, false);
    }

    // D element (reg r, lane) -> row = row0 + r + 8*hi, col = t*16 + lo
#pragma unroll
    for (int r = 0; r < 8; ++r) {
        const size_t rbase = (size_t)(row0 + r + 8 * hi) * D_EN;
        Qout[rbase +  0 + lo] = q0[r];
        Qout[rbase + 16 + lo] = q1[r];
        Qout[rbase + 32 + lo] = q2[r];
        Qout[rbase + 48 + lo] = q3[r];
        Kout[rbase +  0 + lo] = k0[r];
        Kout[rbase + 16 + lo] = k1[r];
        Kout[rbase + 32 + lo] = k2[r];
        Kout[rbase + 48 + lo] = k3[r];
    }
}

// ---------------------------------------------------------------------------
// Kernel 2: charge[row] = sigmoid(dot(features[row], w_charge) + b_charge)
// ---------------------------------------------------------------------------
__global__ void ef_charge_kernel(const float* __restrict__ feats,
                                 const float* __restrict__ w_charge,
                                 const float* __restrict__ b_charge,
                                 float* __restrict__ charge) {
    const int wid  = threadIdx.x >> 5;
    const int lane = threadIdx.x & 31;
    const int row  = blockIdx.x * 8 + wid;

    const float* fRow = feats + (size_t)row * F_DIM;
    float sum = 0.0f;
#pragma unroll
    for (int i = 0; i < F_DIM / 32; ++i) {
        const int c = lane + 32 * i;
        sum = fmaf(fRow[c], w_charge[c], sum);
    }
#pragma unroll
    for (int off = 16; off > 0; off >>= 1)
        sum += __shfl_xor(sum, off, 32);

    if (lane == 0) {
        float x = sum + b_charge[0];
        charge[row] = 1.0f / (1.0f + expf(-x));
    }
}

// ---------------------------------------------------------------------------
// Kernel 3: energy. Block (8 waves) computes 128(i) x 128(j).
// K tile (128x64 f32) is TDM-staged into LDS once per block (one
// tensor_load_to_lds per wave for its 16-row slice, with 2-dword row padding
// so the 64-bank LDS is conflict-light). Each wave then computes a 16x128
// tile: A (its own Q rows) from global, B from LDS, 128 f32 WMMAs.
// ---------------------------------------------------------------------------
__global__ void ef_energy_kernel(const float* __restrict__ Q,        // [B*S,64]
                                 const float* __restrict__ K,        // [B*S,64]
                                 const float* __restrict__ charge,   // [B*S]
                                 const float* __restrict__ ls_ptr,   // [1]
                                 float* __restrict__ out) {          // [B,S,S]
    __shared__ __align__(16) float sK[128 * KROW_PAD];   // 33 KB

    const int lane = threadIdx.x & 31;
    const int wid  = threadIdx.x >> 5;
    const int lo   = lane & 15;
    const int hi   = lane >> 4;

    const int blk = blockIdx.x;          // b*1024 + ib*32 + jb
    const int b   = blk >> 10;
    const int ib  = (blk >> 5) & 31;
    const int jb  = blk & 31;
    const int i0  = ib * 128;
    const int j0  = jb * 128;

    // ---- TDM stage: wave w loads K rows [j0+16w, j0+16w+16) into its slice
    {
        const int ws = __builtin_amdgcn_readfirstlane(wid);   // SGPR-uniform
        unsigned lds_base = (unsigned)(size_t)(void*)&sK[0];
        unsigned lds_addr = lds_base + (unsigned)(ws * 16 * KROW_PAD * 4);
        unsigned long long ga =
            (unsigned long long)(size_t)(K + (size_t)(b * S_LEN + j0 + ws * 16) * D_EN);

        u32x4 g0;
        g0.x = 1u;                                            // count=1
        g0.y = lds_addr;                                      // lds_addr
        g0.z = (unsigned)(ga & 0xFFFFFFFFull);                // global_addr lo
        g0.w = (unsigned)((ga >> 32) & 0x01FFFFFFull)         // global_addr hi
               | 0x80000000u;                                 // type=2 ("image")
        i32x8 g1;
        g1[0] = (int)((2u << 16)        // data_size = 4 bytes
                    | (1u << 20)        // pad_enable
                    | (5u << 22)        // pad_interval: every 64 dwords
                    | (1u << 25));      // pad_amount: 2 dwords
        g1[1] = (int)(64u << 16);       // tensor_dim0 = 64
        g1[2] = (int)(16u << 16);       // tensor_dim1 = 16
        g1[3] = (int)(64u << 16);       // tile_dim0 = 64
        g1[4] = 16;                     // tile_dim1 = 16 rows
        g1[5] = 64;                     // tensor_dim0_stride = 64
        g1[6] = 0;
        g1[7] = 0;
        i32x4 z4 = {0, 0, 0, 0};
#if defined(__clang_major__) && (__clang_major__ >= 23)
        i32x8 z8 = {0, 0, 0, 0, 0, 0, 0, 0};
        __builtin_amdgcn_tensor_load_to_lds(g0, g1, z4, z4, z8, 0);
#else
        __builtin_amdgcn_tensor_load_to_lds(g0, g1, z4, z4, 0);
#endif
        __builtin_amdgcn_s_wait_tensorcnt(0);
    }
    __syncthreads();

    // ---- compute 16x128 per wave
    const float* Qrow = Q + (size_t)(b * S_LEN + i0 + wid * 16 + lo) * D_EN;

    v8f acc[8];
#pragma unroll
    for (int t = 0; t < 8; ++t) acc[t] = (v8f){};

#pragma unroll
    for (int kk = 0; kk < D_EN / 4; ++kk) {
        const int d = kk * 4 + hi * 2;
        v2f a = *(const v2f*)(Qrow + d);
#pragma unroll
        for (int t = 0; t < 8; ++t) {
            v2f bb = *(const v2f*)(&sK[(t * 16 + lo) * KROW_PAD + d]);
            acc[t] = __builtin_amdgcn_wmma_f32_16x16x4_f32(
                         false, a, false, bb, (short)0, acc[t], false, false);
        }
    }

    // ---- fused epilogue: -(QK/8) * ls/dist * c_i * c_j, non-temporal stores
    const float factor = -0.125f * ls_ptr[0];
    const float* chgB = charge + (size_t)b * S_LEN;

    float cj[8];
#pragma unroll
    for (int t = 0; t < 8; ++t) cj[t] = chgB[j0 + t * 16 + lo];

#pragma unroll
    for (int r = 0; r < 8; ++r) {
        const int i  = i0 + wid * 16 + r + 8 * hi;
        const float ci = chgB[i] * factor;
        float* orow = out + ((size_t)b * S_LEN + i) * (size_t)S_LEN;
#pragma unroll
        for (int t = 0; t < 8; ++t) {
            const int j = j0 + t * 16 + lo;
            float dd = fmaxf(fabsf((float)(i - j)), 1.0f);
            __builtin_nontemporal_store(acc[t][r] * ci * cj[t] / dd, orow + j);
        }
    }
}

// ---------------------------------------------------------------------------
extern "C" void kernel_launch(void* const* d_in, const int* in_sizes, int n_in,
                              void* d_out, int out_size, void* d_ws, size_t ws_size,
                              hipStream_t stream) {
    const float* feats    = (const float*)d_in[0];   // [4,4096,512]
    const float* Wq       = (const float*)d_in[1];   // [512,64]
    const float* Wk       = (const float*)d_in[2];   // [512,64]
    const float* w_charge = (const float*)d_in[3];   // [512]
    const float* b_charge = (const float*)d_in[4];   // [1]
    const float* ls       = (const float*)d_in[5];   // [1]
    float* out = (float*)d_out;                      // [4,4096,4096]

    const int nRows = in_sizes[0] / F_DIM;           // B*S = 16384
    const int nB    = nRows / S_LEN;                 // 4

    // Workspace carve-up (floats)
    float* Qbuf = (float*)d_ws;                         // nRows*64   = 4 MB
    float* Kbuf = Qbuf + (size_t)nRows * D_EN;          //            = 4 MB
    float* Cbuf = Kbuf + (size_t)nRows * D_EN;          // nRows      = 64 KB
    float* WqT  = Cbuf + (size_t)nRows;                 // 64*512     = 128 KB
    float* WkT  = WqT + (size_t)D_EN * F_DIM;           // 64*512     = 128 KB

    // 0) transpose weights
    ef_transpose_kernel<<<(F_DIM * D_EN) / 256, 256, 0, stream>>>(Wq, Wk, WqT, WkT);

    // 1) projections: nRows/16 row-tiles, 8 waves per block
    ef_proj_kernel<<<(nRows / 16) / 8, 256, 0, stream>>>(feats, WqT, WkT, Qbuf, Kbuf);

    // 2) charge: wave per row
    ef_charge_kernel<<<nRows / 8, 256, 0, stream>>>(feats, w_charge, b_charge, Cbuf);

    // 3) energy: block = 128x128 output tile, 8 waves
    ef_energy_kernel<<<nB * 32 * 32, 256, 0, stream>>>(Qbuf, Kbuf, Cbuf, ls, out);
}